// GlobalModel_28346784153767
// MI455X (gfx1250) — compile-verified
//
#include <hip/hip_runtime.h>
#include <hip/hip_bf16.h>

// ---------------------------------------------------------------------------
// GlobalModel: segment-mean pooling (sorted batch) + 2-layer MLP.
// Phase 1 is HBM-bound (512 MB stream ~22us @ 23.3 TB/s) -> register-
// accumulated segment sums with atomics only at segment boundaries.
// Phase 2 GEMMs ([4096,256]x[256,256], [4096,256]x[256,128]) are tiny and
// L2-resident -> native f32 WMMA (v_wmma_f32_16x16x4_f32), one wave per
// 16x16 output tile, full f32 precision matching the reference.
// ---------------------------------------------------------------------------

typedef __attribute__((ext_vector_type(2))) float v2f;
typedef __attribute__((ext_vector_type(8))) float v8f;

#define DDIM 128
#define SEG_BLOCKS 1024

__global__ void zero_kernel(float* __restrict__ p, int n) {
    int i = blockIdx.x * blockDim.x + threadIdx.x;
    if (i < n) p[i] = 0.0f;
}

// One thread per feature column (blockDim.x == 128). Each block walks a
// contiguous chunk of rows; since `batch` is sorted, we accumulate in
// registers and flush with atomics only when the segment id changes.
__global__ void seg_accum(const float* __restrict__ x,
                          const float* __restrict__ ea,
                          const int* __restrict__ batch,
                          float* __restrict__ node_sum,
                          float* __restrict__ edge_sum,
                          float* __restrict__ cnt,
                          int nRows, int rowsPerBlock) {
    int d  = threadIdx.x;                       // 0..127
    int r0 = blockIdx.x * rowsPerBlock;
    int r1 = r0 + rowsPerBlock;
    if (r1 > nRows) r1 = nRows;
    if (r0 >= r1) return;

    float ax = 0.0f, ae = 0.0f;
    int bcur = batch[r0];
    int run  = 0;
    for (int r = r0; r < r1; ++r) {
        int b = batch[r];
        if (b != bcur) {
            atomicAdd(&node_sum[bcur * DDIM + d], ax);
            atomicAdd(&edge_sum[bcur * DDIM + d], ae);
            if (d == 0) atomicAdd(&cnt[bcur], (float)run);
            ax = 0.0f; ae = 0.0f; run = 0; bcur = b;
        }
        size_t base = (size_t)r * DDIM + d;
        ax += x[base];
        ae += ea[base];
        ++run;
    }
    atomicAdd(&node_sum[bcur * DDIM + d], ax);
    atomicAdd(&edge_sum[bcur * DDIM + d], ae);
    if (d == 0) atomicAdd(&cnt[bcur], (float)run);
}

// Build [B, 2D] = [node_mean | edge_mean]
__global__ void build_out(const float* __restrict__ node_sum,
                          const float* __restrict__ edge_sum,
                          const float* __restrict__ cnt,
                          float* __restrict__ outm, int nB) {
    int idx = blockIdx.x * blockDim.x + threadIdx.x;
    int total = nB * (2 * DDIM);
    if (idx >= total) return;
    int b = idx >> 8;          // / 256
    int j = idx & 255;
    float denom = cnt[b] + 1e-6f;
    float v = (j < DDIM) ? node_sum[b * DDIM + j]
                         : edge_sum[b * DDIM + (j - DDIM)];
    outm[idx] = v / denom;
}

// C[M,N] = act(A[M,K] @ B[K,N] + bias[N]); row-major; one wave per 16x16 tile.
// V_WMMA_F32_16X16X4_F32 operand layout (ISA 7.12.2):
//   A 16x4 : lane%16 = M row; VGPR0 holds K = 2*(lane/16), VGPR1 = K+1
//   B 4x16 : lane%16 = N col; VGPR0 holds K = 2*(lane/16), VGPR1 = K+1
//   C 16x16: VGPR r -> M = r + 8*(lane/16), N = lane%16
__global__ void wmma_gemm_f32(const float* __restrict__ A,
                              const float* __restrict__ Bm,
                              const float* __restrict__ bias,
                              float* __restrict__ C,
                              int M, int N, int K, int doRelu) {
    int wavesPerBlock = blockDim.x >> 5;
    int wave = blockIdx.x * wavesPerBlock + (threadIdx.x >> 5);
    int tilesM = M >> 4, tilesN = N >> 4;
    if (wave >= tilesM * tilesN) return;       // uniform per-wave exit
    int mTile = wave / tilesN;
    int nTile = wave % tilesN;

    int lane = threadIdx.x & 31;
    int r    = lane & 15;                      // A row / B,C col within tile
    int half = lane >> 4;                      // 0 or 1
    int m0 = mTile << 4, n0 = nTile << 4;

    const float* aPtr = A + (size_t)(m0 + r) * K + 2 * half; // float2 aligned
    const float* bPtr = Bm + (size_t)(2 * half) * N + n0 + r;

    v8f acc = {};
    for (int k0 = 0; k0 < K; k0 += 4) {
        v2f a = *reinterpret_cast<const v2f*>(aPtr + k0);
        v2f b;
        b.x = bPtr[(size_t)k0 * N];
        b.y = bPtr[(size_t)k0 * N + N];
        acc = __builtin_amdgcn_wmma_f32_16x16x4_f32(
            /*neg_a=*/false, a, /*neg_b=*/false, b,
            /*c_mod=*/(short)0, acc, /*reuse_a=*/false, /*reuse_b=*/false);
    }

    float bv = bias[n0 + r];
    int mBase = m0 + half * 8;
    #pragma unroll
    for (int i = 0; i < 8; ++i) {
        float v = acc[i] + bv;
        if (doRelu) v = fmaxf(v, 0.0f);
        C[(size_t)(mBase + i) * N + n0 + r] = v;
    }
}

extern "C" void kernel_launch(void* const* d_in, const int* in_sizes, int n_in,
                              void* d_out, int out_size, void* d_ws, size_t ws_size,
                              hipStream_t stream) {
    const float* x   = (const float*)d_in[0];   // [N, 128]
    const float* ea  = (const float*)d_in[1];   // [N, 128]
    /* d_in[2] = u, only shape used */
    const float* W1  = (const float*)d_in[3];   // [256, H]
    const float* b1  = (const float*)d_in[4];   // [H]
    const float* W2  = (const float*)d_in[5];   // [H, 128]
    const float* b2  = (const float*)d_in[6];   // [128]
    /* d_in[7] = edge_index: unused by reference */
    const int* batch = (const int*)d_in[8];     // [N], sorted
    float* out = (float*)d_out;                 // [B, 128]

    const int D  = DDIM;
    const int N  = in_sizes[0] / D;             // 500000
    const int Bn = in_sizes[2] / D;             // 4096
    const int H  = in_sizes[4];                 // 256

    // Workspace layout (floats)
    float* ws       = (float*)d_ws;
    float* node_sum = ws;                           // B*D
    float* edge_sum = node_sum + (size_t)Bn * D;    // B*D
    float* cnt      = edge_sum + (size_t)Bn * D;    // B
    float* outm     = cnt + Bn;                     // B*2D
    float* h        = outm + (size_t)Bn * 2 * D;    // B*H

    // 1) zero accumulators
    int zn = 2 * Bn * D + Bn;
    zero_kernel<<<(zn + 255) / 256, 256, 0, stream>>>(ws, zn);

    // 2) segment sums (sorted batch -> register accumulation, boundary atomics)
    int rpb = (N + SEG_BLOCKS - 1) / SEG_BLOCKS;
    seg_accum<<<SEG_BLOCKS, DDIM, 0, stream>>>(x, ea, batch, node_sum, edge_sum,
                                               cnt, N, rpb);

    // 3) means + concat -> [B, 2D]
    int totalO = Bn * 2 * D;
    build_out<<<(totalO + 255) / 256, 256, 0, stream>>>(node_sum, edge_sum, cnt,
                                                        outm, Bn);

    // 4) h = relu(outm @ W1 + b1)   [Bn,2D] x [2D,H]
    {
        int M = Bn, Nn = H, K = 2 * D;
        int waves = (M / 16) * (Nn / 16);
        wmma_gemm_f32<<<(waves + 3) / 4, 128, 0, stream>>>(outm, W1, b1, h,
                                                           M, Nn, K, 1);
    }
    // 5) out = h @ W2 + b2          [Bn,H] x [H,D]
    {
        int M = Bn, Nn = D, K = H;
        int waves = (M / 16) * (Nn / 16);
        wmma_gemm_f32<<<(waves + 3) / 4, 128, 0, stream>>>(h, W2, b2, out,
                                                           M, Nn, K, 0);
    }
}